// DiverseSiblingsSearch_73744588472778
// MI455X (gfx1250) — compile-verified
//
#include <hip/hip_runtime.h>
#include <stdint.h>

// ---------------------------------------------------------------------------
// DiverseSiblingsSearch for MI455X (gfx1250).
//   lprobs: (128, 5, 50257) f32, scores: (128, 5, 10) f32, step scalar int.
//   Out: final_scores(128,10) ++ final_indices(128,10) ++ final_beams(128,10)
//        all as f32, concatenated flat.
// Memory-bound: 128.6 MB streamed once -> ~5.5us floor @ 23.3 TB/s.
// Kernel 1: one block per (b,beam) row. TDM ring-buffered tile DMA to LDS
//           (wave 0 issues, lookahead 2, S_WAIT_TENSORCNT), 256 threads keep
//           register top-10 lists fed by ds_load_b128, then LDS tournament
//           merge of sorted 10-lists.
// Kernel 2: per-batch top-10 of the 5*10 penalized candidates.
// ---------------------------------------------------------------------------

#define BSZ     128
#define BEAM    5
#define VOCAB   50257
#define SSTEP   10
#define K       10
#define NROWS   (BSZ * BEAM)
#define THREADS 256
#define TILE    4096            // floats per TDM tile (16 KB)
#define NBUF    4               // ring depth (lookahead 2 + read lag 1 + 1)
#define DIVERSITY_RATE 0.5f

#if defined(__gfx1250__) && defined(__has_builtin)
#if __has_builtin(__builtin_amdgcn_tensor_load_to_lds) && __has_builtin(__builtin_amdgcn_s_wait_tensorcnt)
#define USE_TDM 1
#endif
#endif
#ifndef USE_TDM
#define USE_TDM 0
#endif

typedef uint32_t v4u __attribute__((ext_vector_type(4)));
typedef int      v8i __attribute__((ext_vector_type(8)));
typedef int      v4i __attribute__((ext_vector_type(4)));

#if USE_TDM
// Build a D# for a 1-row tile of `cnt` f32 elements: global -> LDS.
// Group0: count=1 | lds_addr | global_addr | type=2("image").
// Group1: data_size=4B, tensor_dim0=cnt, tensor_dim1=1, tile_dim0=cnt,
//         tile_dim1=1, tensor_dim0_stride=cnt.  Groups 2/3 + trailing
// group zero (tensor is <=2D).  6-arg builtin: (v4u, v8i, v4i, v4i, v8i, cpol).
__device__ __forceinline__ void tdm_load_tile(const float* gsrc,
                                              uint32_t lds_addr, int cnt) {
  uint64_t ga = (uint64_t)(uintptr_t)gsrc;
  v4u g0;
  g0.x = 1u;                                             // count=1, user mode
  g0.y = lds_addr;                                       // LDS byte address
  g0.z = (uint32_t)ga;                                   // global_addr[31:0]
  g0.w = (uint32_t)((ga >> 32) & 0x01FFFFFFu) | (2u << 30); // [56:32] | type=2
  uint32_t c = (uint32_t)cnt;
  v8i g1;
  g1[0] = (int)(2u << 16);                 // workgroup_mask=0, data_size=2 (4B)
  g1[1] = (int)((c & 0xFFFFu) << 16);      // [63:48] = tensor_dim0[15:0]
  g1[2] = (int)((c >> 16) | (1u << 16));   // tensor_dim0[31:16] | tensor_dim1=1
  g1[3] = (int)((c & 0xFFFFu) << 16);      // tensor_dim1 hi=0 | tile_dim0=cnt
  g1[4] = 1;                               // tile_dim1=1, tile_dim2=0
  g1[5] = (int)c;                          // tensor_dim0_stride[31:0]
  g1[6] = 0;                               // stride hi | tensor_dim1_stride lo
  g1[7] = 0;
  v4i g2 = {0, 0, 0, 0};
  v4i g3 = {0, 0, 0, 0};
  v8i g4 = {0, 0, 0, 0, 0, 0, 0, 0};
  __builtin_amdgcn_tensor_load_to_lds(g0, g1, g2, g3, g4, 0);
}
#endif

__global__ __launch_bounds__(THREADS)
void ds_row_topk(const float* __restrict__ lprobs,
                 const float* __restrict__ scores,
                 const int*   __restrict__ stepPtr,
                 float* __restrict__ wsVal,
                 int*   __restrict__ wsIdx) {
  __shared__ float sval[THREADS * K];
  __shared__ int   sidx[THREADS * K];
#if USE_TDM
  __shared__ float tileBuf[NBUF][TILE];
#endif
  const int row = blockIdx.x;            // row = b*BEAM + beam
  const int tid = threadIdx.x;
  const float* rowPtr = lprobs + (size_t)row * VOCAB;

  const float NEG = -__builtin_huge_valf();
  float v[K];
  int   id[K];
#pragma unroll
  for (int r = 0; r < K; ++r) { v[r] = NEG; id[r] = 0; }

  auto insert = [&](float val, int gi) {
    if (val > v[K - 1]) {                // beats current min of top-10
      v[K - 1] = val; id[K - 1] = gi;
#pragma unroll
      for (int r = K - 1; r > 0; --r) {  // bubble up (strict: ties keep order)
        if (v[r] > v[r - 1]) {
          float tv = v[r]; v[r] = v[r - 1]; v[r - 1] = tv;
          int   ti = id[r]; id[r] = id[r - 1]; id[r - 1] = ti;
        }
      }
    }
  };

#if USE_TDM
  // ---- TDM ring-buffered streaming, wave 0 drives the DMA, lookahead 2 ----
  const int nTiles = (VOCAB + TILE - 1) / TILE;
  const bool issuer = (tid < 32);
  if (issuer) {
#pragma unroll
    for (int p = 0; p < 2; ++p) {        // prime the pipeline (nTiles >= 2)
      const int base = p * TILE;
      const int rem  = VOCAB - base;
      tdm_load_tile(rowPtr + base, (uint32_t)(uintptr_t)&tileBuf[p][0],
                    rem < TILE ? rem : TILE);
    }
  }
  for (int t = 0; t < nTiles; ++t) {
    const int base = t * TILE;
    const int rem  = VOCAB - base;
    const int cnt  = (rem < TILE) ? rem : TILE;
    if (issuer) {
      if (t + 2 < nTiles) {              // keep 3 tiles in flight
        const int b2 = base + 2 * TILE;
        const int r2 = VOCAB - b2;
        tdm_load_tile(rowPtr + b2, (uint32_t)(uintptr_t)&tileBuf[(t + 2) % NBUF][0],
                      r2 < TILE ? r2 : TILE);
        __builtin_amdgcn_s_wait_tensorcnt(2);   // tile t landed (in-order)
      } else if (t + 2 == nTiles) {
        __builtin_amdgcn_s_wait_tensorcnt(1);
      } else {
        __builtin_amdgcn_s_wait_tensorcnt(0);
      }
    }
    __syncthreads();   // publish tile t; slowest reader is on tile t-1, and
                       // in-flight DMA targets t..t+2 -> 4 distinct buffers
    const float*  buf  = tileBuf[t % NBUF];
    const float4* buf4 = reinterpret_cast<const float4*>(buf);
    const int cnt4 = cnt >> 2;
    for (int j = tid; j < cnt4; j += THREADS) {     // ds_load_b128 per 4 elems
      float4 x = buf4[j];
      const int gi = base + j * 4;
      insert(x.x, gi); insert(x.y, gi + 1); insert(x.z, gi + 2); insert(x.w, gi + 3);
    }
    for (int j = (cnt4 << 2) + tid; j < cnt; j += THREADS)   // tail (<4 elems)
      insert(buf[j], base + j);
  }
  __syncthreads();     // last tile's readers done before merge reuses nothing,
                       // but keeps phases cleanly separated
#else
  // ---- Fallback: direct coalesced b128 streaming loads ----
  const int n4 = VOCAB / 4;
  const float4* rp4 = reinterpret_cast<const float4*>(rowPtr);
  for (int j = tid; j < n4; j += THREADS) {
    float4 x = rp4[j];
    const int gi = j * 4;
    insert(x.x, gi); insert(x.y, gi + 1); insert(x.z, gi + 2); insert(x.w, gi + 3);
  }
  for (int j = n4 * 4 + tid; j < VOCAB; j += THREADS) insert(rowPtr[j], j);
#endif

  // ---- tournament merge of 256 sorted 10-lists -> block top-10 ----
#pragma unroll
  for (int r = 0; r < K; ++r) { sval[tid * K + r] = v[r]; sidx[tid * K + r] = id[r]; }
  __syncthreads();
  for (int stride = THREADS / 2; stride >= 1; stride >>= 1) {
    if (tid < stride) {
      const int A = tid * K, B = (tid + stride) * K;
      float ov[K]; int oi[K];
      int i = 0, j = 0;
#pragma unroll
      for (int r = 0; r < K; ++r) {      // i+j==r<=9 -> no OOB reads
        float a = sval[A + i], b = sval[B + j];
        if (a >= b) { ov[r] = a; oi[r] = sidx[A + i]; ++i; }
        else        { ov[r] = b; oi[r] = sidx[B + j]; ++j; }
      }
#pragma unroll
      for (int r = 0; r < K; ++r) { sval[A + r] = ov[r]; sidx[A + r] = oi[r]; }
    }
    __syncthreads();
  }

  if (tid == 0) {
    const int step = *stepPtr;
    const float sc = scores[row * SSTEP + (step - 1)];
#pragma unroll
    for (int r = 0; r < K; ++r) {        // add running score, rank penalty
      wsVal[row * K + r] = sval[r] + sc - (float)(r + 1) * DIVERSITY_RATE;
      wsIdx[row * K + r] = sidx[r];
    }
  }
}

__global__ void ds_final_topk(const float* __restrict__ wsVal,
                              const int*   __restrict__ wsIdx,
                              float* __restrict__ out) {
  const int b = blockIdx.x * blockDim.x + threadIdx.x;
  if (b >= BSZ) return;
  const float NEG = -__builtin_huge_valf();
  float v[K]; int vi[K]; int vb[K];
#pragma unroll
  for (int r = 0; r < K; ++r) { v[r] = NEG; vi[r] = 0; vb[r] = 0; }
  const float* base  = wsVal + (size_t)b * BEAM * K;
  const int*   ibase = wsIdx + (size_t)b * BEAM * K;
  for (int c = 0; c < BEAM * K; ++c) {   // flat order == top_k tie order
    const float val = base[c];
    if (val > v[K - 1]) {
      v[K - 1] = val; vi[K - 1] = ibase[c]; vb[K - 1] = c / K;
#pragma unroll
      for (int r = K - 1; r > 0; --r) {
        if (v[r] > v[r - 1]) {
          float tv = v[r]; v[r] = v[r - 1]; v[r - 1] = tv;
          int t0 = vi[r]; vi[r] = vi[r - 1]; vi[r - 1] = t0;
          int t1 = vb[r]; vb[r] = vb[r - 1]; vb[r - 1] = t1;
        }
      }
    }
  }
#pragma unroll
  for (int r = 0; r < K; ++r) {
    out[b * K + r]                 = v[r];
    out[BSZ * K + b * K + r]       = (float)vi[r];
    out[2 * BSZ * K + b * K + r]   = (float)vb[r];
  }
}

extern "C" void kernel_launch(void* const* d_in, const int* in_sizes, int n_in,
                              void* d_out, int out_size, void* d_ws, size_t ws_size,
                              hipStream_t stream) {
  const float* lprobs = (const float*)d_in[0];
  const float* scores = (const float*)d_in[1];
  const int*   step   = (const int*)d_in[2];
  (void)in_sizes; (void)n_in; (void)out_size; (void)ws_size;

  float* wsVal = (float*)d_ws;                       // NROWS*K floats
  int*   wsIdx = (int*)((char*)d_ws + NROWS * K * sizeof(float));

  ds_row_topk<<<NROWS, THREADS, 0, stream>>>(lprobs, scores, step, wsVal, wsIdx);
  ds_final_topk<<<(BSZ + 63) / 64, 64, 0, stream>>>(wsVal, wsIdx, (float*)d_out);
}